// OuterProductMHSA_14181982012035
// MI455X (gfx1250) — compile-verified
//
#include <hip/hip_runtime.h>

#define NEG_BIG (-1.0e9f)

// Problem dims (fixed by the reference)
#define BB   4
#define NN   384
#define DD   256
#define MM   384
#define ROWS (BB * NN)   // 1536 flattened GEMM rows

#define TN   8           // n-rows per block in phase 2 (K/V reuse factor)
#define MHALF (MM / 2)   // m-range split across the two half-blocks

typedef __attribute__((ext_vector_type(2))) float v2f;
typedef __attribute__((ext_vector_type(8))) float v8f;

__device__ __forceinline__ float fast_tanh(float x) {
#if __has_builtin(__builtin_amdgcn_tanhf)
    return __builtin_amdgcn_tanhf(x);   // v_tanh_f32 on gfx1250
#else
    return tanhf(x);
#endif
}

// ---------------------------------------------------------------------------
// Phase 1: three fused projections  O_g = A_g @ W_g + b_g   (g = q,k,v)
// One wave per 16x16 C tile, V_WMMA_F32_16X16X4_F32 stepping K by 4.
// (Verified in round 1: emits v_wmma_f32_16x16x4_f32.)
// ---------------------------------------------------------------------------
__global__ __launch_bounds__(256) void proj_gemm_wmma(
    const float* __restrict__ A0, const float* __restrict__ A1, const float* __restrict__ A2,
    const float* __restrict__ W0, const float* __restrict__ W1, const float* __restrict__ W2,
    const float* __restrict__ b0, const float* __restrict__ b1, const float* __restrict__ b2,
    float* __restrict__ O0, float* __restrict__ O1, float* __restrict__ O2)
{
    const int TILES_Nd      = DD / 16;                 // 16
    const int TILES_M       = ROWS / 16;               // 96
    const int TILES_PER_MAT = TILES_M * TILES_Nd;      // 1536

    const int lane = threadIdx.x & 31;
    const int wave = blockIdx.x * (blockDim.x >> 5) + (threadIdx.x >> 5);
    const int g    = wave / TILES_PER_MAT;             // 0..2: which projection
    const int t    = wave - g * TILES_PER_MAT;
    const int tm   = t / TILES_Nd;
    const int tn   = t - tm * TILES_Nd;

    const float* __restrict__ A    = (g == 0) ? A0 : (g == 1) ? A1 : A2;
    const float* __restrict__ W    = (g == 0) ? W0 : (g == 1) ? W1 : W2;
    const float* __restrict__ bias = (g == 0) ? b0 : (g == 1) ? b1 : b2;
    float*       __restrict__ O    = (g == 0) ? O0 : (g == 1) ? O1 : O2;

    const int arow  = tm * 16 + (lane & 15);
    const int khalf = (lane >> 4) << 1;                // 0 or 2
    const int ncol  = tn * 16 + (lane & 15);
    const int rbase = tm * 16 + ((lane >> 4) << 3);    // C row base for this lane half

    v8f c = {};

#if __has_builtin(__builtin_amdgcn_wmma_f32_16x16x4_f32)
    const float* __restrict__ arowp = A + (size_t)arow * DD;
    for (int k = 0; k < DD; k += 4) {
        v2f a, b;
        a.x = arowp[k + khalf];
        a.y = arowp[k + khalf + 1];
        b.x = W[(size_t)(k + khalf)     * DD + ncol];
        b.y = W[(size_t)(k + khalf + 1) * DD + ncol];
        // (neg_a, A, neg_b, B, c_mod, C, reuse_a, reuse_b)
        c = __builtin_amdgcn_wmma_f32_16x16x4_f32(false, a, false, b,
                                                  (short)0, c, false, false);
    }
#else
    for (int r = 0; r < 8; ++r) {
        float acc = 0.0f;
        for (int k = 0; k < DD; ++k)
            acc = fmaf(A[(size_t)(rbase + r) * DD + k], W[(size_t)k * DD + ncol], acc);
        c[r] = acc;
    }
#endif

    const float bv = bias[ncol];
#pragma unroll
    for (int r = 0; r < 8; ++r)
        O[(size_t)(rbase + r) * DD + ncol] = c[r] + bv;
}

// ---------------------------------------------------------------------------
// Phase 2: out[b,n,d] = sum_m tanh(Q[b,n,d]*K[b,m,d] + mask[b,n,m]*NEG_BIG) * V[b,m,d]
//
// Register-tiled over TN=8 rows of n: each K/V element loaded once per block
// feeds 8 tanh/FMA chains -> L2 traffic drops 8x (1.2 GB -> 150 MB), making
// the kernel tanh-throughput bound (the irreducible cost).
//
// 512 threads: d = tid & 255 picks the channel (coalesced 1 KB row loads),
// tid >> 8 picks which half of the m-range; halves combine through LDS in a
// fixed order (deterministic, no atomics). 192 blocks x 16 waves = 3072 waves.
// Mask bias stored as sbias[m][8] so per-iteration bias reads are two
// broadcast ds_load_b128 instead of 8 scalar DS ops.
// ---------------------------------------------------------------------------
__global__ __launch_bounds__(512) void op_mhsa_main(
    const float* __restrict__ Qp, const float* __restrict__ Kp,
    const float* __restrict__ Vp, const int* __restrict__ mask,
    float* __restrict__ out)
{
    const int gb   = blockIdx.x;            // 0 .. B*(N/TN)-1 = 191
    const int b    = gb / (NN / TN);
    const int ng   = gb - b * (NN / TN);
    const int n0   = ng * TN;
    const int d    = threadIdx.x & (DD - 1);
    const int half = threadIdx.x >> 8;      // 0 or 1: m-range half

    __shared__ __align__(16) float sbias[MM][TN];   // 12 KB
    __shared__ float spart[TN][DD];                 //  8 KB

    const int* __restrict__ mbase = mask + ((size_t)b * NN + n0) * MM;
    for (int i = threadIdx.x; i < MM * TN; i += 512) {
        const int m = i >> 3, j = i & 7;
        sbias[m][j] = (float)mbase[(size_t)j * MM + m] * NEG_BIG;
    }
    __syncthreads();

    float qv[TN], acc[TN];
#pragma unroll
    for (int j = 0; j < TN; ++j) {
        qv[j]  = Qp[(size_t)(b * NN + n0 + j) * DD + d];
        acc[j] = 0.0f;
    }

    const int m0 = half * MHALF;
    const int m1 = m0 + MHALF;
    const float* __restrict__ Kb = Kp + (size_t)b * NN * DD + d;
    const float* __restrict__ Vb = Vp + (size_t)b * NN * DD + d;

#pragma unroll 2
    for (int m = m0; m < m1; ++m) {
        const float kv = Kb[(size_t)m * DD];
        const float vv = Vb[(size_t)m * DD];
        const float4* sb = (const float4*)(&sbias[m][0]);
        const float4 bl = sb[0], bh = sb[1];
        const float bb[TN] = {bl.x, bl.y, bl.z, bl.w, bh.x, bh.y, bh.z, bh.w};
#pragma unroll
        for (int j = 0; j < TN; ++j) {
            const float s = fmaf(qv[j], kv, bb[j]);
            acc[j] = fmaf(fast_tanh(s), vv, acc[j]);
        }
    }

    // Deterministic two-way combine: half 1 parks partials in LDS,
    // half 0 adds them in a fixed order and stores.
    if (half == 1) {
#pragma unroll
        for (int j = 0; j < TN; ++j) spart[j][d] = acc[j];
    }
    __syncthreads();
    if (half == 0) {
#pragma unroll
        for (int j = 0; j < TN; ++j)
            out[(size_t)(b * NN + n0 + j) * DD + d] = acc[j] + spart[j][d];
    }
}

// ---------------------------------------------------------------------------
extern "C" void kernel_launch(void* const* d_in, const int* in_sizes, int n_in,
                              void* d_out, int out_size, void* d_ws, size_t ws_size,
                              hipStream_t stream) {
    const float* q    = (const float*)d_in[0];
    const float* k    = (const float*)d_in[1];
    const float* v    = (const float*)d_in[2];
    const int*   mask = (const int*)  d_in[3];
    const float* Wq   = (const float*)d_in[4];
    const float* bq   = (const float*)d_in[5];
    const float* Wk   = (const float*)d_in[6];
    const float* bk   = (const float*)d_in[7];
    const float* Wv   = (const float*)d_in[8];
    const float* bv   = (const float*)d_in[9];
    float* out = (float*)d_out;

    // Workspace: Q', K', V' projections (3 * 1536*256 f32 = 4.72 MB)
    float* Qp = (float*)d_ws;
    float* Kp = Qp + (size_t)ROWS * DD;
    float* Vp = Kp + (size_t)ROWS * DD;

    // 3 matrices * 1536 tiles = 4608 waves; 8 waves (256 thr) per block.
    proj_gemm_wmma<<<576, 256, 0, stream>>>(q, k, v, Wq, Wk, Wv,
                                            bq, bk, bv, Qp, Kp, Vp);

    // B * N/TN = 192 blocks, 512 threads (16 waves) each.
    op_mhsa_main<<<BB * (NN / TN), 512, 0, stream>>>(Qp, Kp, Vp, mask, out);
}